// VQEmbedding_85753317032646
// MI455X (gfx1250) — compile-verified
//
#include <hip/hip_runtime.h>
#include <float.h>
#include <stdint.h>

// ---------------------------------------------------------------------------
// VQ codebook argmin:  idx[n] = argmin_k ( ||e_k||^2 - 2 * z_n . e_k )
//   z: [N=32768, D=256] f32,  e: [K=1024, D=256] f32
// fp32 WMMA (v_wmma_f32_16x16x4_f32) GEMM core, async global->LDS staging
// (global_load_async_to_lds_b64 + s_wait_asynccnt) with double-buffered B.
// ---------------------------------------------------------------------------

typedef __attribute__((ext_vector_type(2))) float v2f;
typedef __attribute__((ext_vector_type(8))) float v8f;

#define D_DIM 256
#define K_DIM 1024
#define THREADS 128                  // 4 wave32
#define WAVES 4
#define STEPS (D_DIM / 4)            // 64 WMMA K-steps over D
#define A_TILE (STEPS * 64)          // 4096 floats per 16-row fragment tile
#define A_FLOATS (WAVES * A_TILE)    // 16384 floats = 64KB
#define B_BUF (STEPS * 128)          // 8192 floats = 32KB per kb tile
                                     //   (chunks interleaved: s*128+chunk*64)
#define B_BASE A_FLOATS
#define LDS_FLOATS (A_FLOATS + 2 * B_BUF)    // 32768 floats = 128KB

// ---- CDNA5 async global->LDS copy (8B per lane), ASYNCcnt-tracked ---------
__device__ __forceinline__ void async_copy_b64(unsigned lds_byte_addr,
                                               const float* gsrc) {
  asm volatile("global_load_async_to_lds_b64 %0, %1, off"
               :: "v"(lds_byte_addr), "v"(gsrc)
               : "memory");
}
__device__ __forceinline__ void wait_async0() {
  asm volatile("s_wait_asynccnt 0x0" ::: "memory");
}
__device__ __forceinline__ unsigned lds_addr_of(const float* p) {
  // flat LDS addresses carry the LDS offset in the low 32 bits
  return (unsigned)(uintptr_t)p;
}

// ---- kernel 1: ||e_k||^2 per code ----------------------------------------
__global__ void __launch_bounds__(THREADS)
vq_enorm_kernel(const float* __restrict__ emb, float* __restrict__ enorm) {
  const int wave = threadIdx.x >> 5;
  const int lane = threadIdx.x & 31;
  const int code = blockIdx.x * WAVES + wave;
  const float* row = emb + (size_t)code * D_DIM;
  float s = 0.f;
#pragma unroll
  for (int i = 0; i < D_DIM / 32; ++i) {
    float v = row[lane + 32 * i];
    s = fmaf(v, v, s);
  }
#pragma unroll
  for (int m = 16; m >= 1; m >>= 1) s += __shfl_xor(s, m, 32);
  if (lane == 0) enorm[code] = s;
}

// ---- async staging of one 32-code B tile for iteration `kb` ---------------
// Fragment (s, chunk, L) lives at base + s*128 + chunk*64 + L*2.
// Each wave-instruction has fixed (chunk, s) and L = lane, so LDS writes
// tile all 64 banks (conflict-free). Strength-reduced: per j the global
// pointer advances 64B and the LDS address 2048B (s += 4).
__device__ __forceinline__ void stage_B_async(float* smem,
                                              const float* __restrict__ emb,
                                              int kb, int buf,
                                              int wave, int lane) {
  const int rsel = lane & 15;          // code within 16-code chunk
  const int csel = (lane >> 4) << 1;   // K-half column offset (0 or 2)
  const int base = B_BASE + buf * B_BUF + wave * 128 + lane * 2;
#pragma unroll
  for (int chunk = 0; chunk < 2; ++chunk) {
    const float* gsrc = emb + (size_t)(kb * 32 + chunk * 16 + rsel) * D_DIM
                            + 4 * wave + csel;
    unsigned lds = lds_addr_of(smem + base + chunk * 64);
#pragma unroll
    for (int j = 0; j < 16; ++j) {
      async_copy_b64(lds, gsrc);
      gsrc += 16;        //  64B: next column group of this code row
      lds  += 2048;      // s += 4 -> 4*128 floats
    }
  }
}

// ---- kernel 2: fused GEMM + argmin ---------------------------------------
__global__ void __launch_bounds__(THREADS)
vq_argmin_kernel(const float* __restrict__ z, const float* __restrict__ emb,
                 const float* __restrict__ enorm, float* __restrict__ out) {
  extern __shared__ float smem[];
  const int tid  = threadIdx.x;
  const int lane = tid & 31;
  const int wave = tid >> 5;
  const int row0 = blockIdx.x * 64;          // 64 query rows per WG

  // ---- stage A (64 rows x 256 f32) async into fragment-permuted LDS ----
  // A fragment (tile, s, L) at tile*4096 + s*64 + L*2 ; s = wave + 4j.
  {
    const int rsel = lane & 15;
    const int csel = (lane >> 4) << 1;
    const int baseA = wave * 64 + lane * 2;
#pragma unroll
    for (int tile = 0; tile < 4; ++tile) {
      const float* gsrc = z + (size_t)(row0 + tile * 16 + rsel) * D_DIM
                            + 4 * wave + csel;
      unsigned lds = lds_addr_of(smem + tile * A_TILE + baseA);
#pragma unroll
      for (int j = 0; j < 16; ++j) {
        async_copy_b64(lds, gsrc);
        gsrc += 16;      //  64B
        lds  += 1024;    // s += 4 -> 4*64 floats
      }
    }
  }
  // ---- stage first B tile async ----
  stage_B_async(smem, emb, 0, 0, wave, lane);
  wait_async0();
  __syncthreads();

  const int aBase = wave * A_TILE + lane * 2;

  float minv[8];
  int   mini[8];
#pragma unroll
  for (int j = 0; j < 8; ++j) { minv[j] = FLT_MAX; mini[j] = 0; }

  for (int kb = 0; kb < K_DIM / 32; ++kb) {
    const int buf = kb & 1;

    // prefetch next tile into the other buffer while we compute
    if (kb + 1 < K_DIM / 32) stage_B_async(smem, emb, kb + 1, buf ^ 1, wave, lane);

    const float* ap = smem + aBase;
    const float* bp = smem + B_BASE + buf * B_BUF + lane * 2;

    // two independent WMMA accumulator chains, software-pipelined by 1 step
    v8f c0 = {};
    v8f c1 = {};
    v2f a  = *(const v2f*)(ap);
    v2f b0 = *(const v2f*)(bp);
    v2f b1 = *(const v2f*)(bp + 64);
#pragma unroll 9
    for (int s = 1; s < STEPS; ++s) {
      v2f an  = *(const v2f*)(ap + s * 64);
      v2f b0n = *(const v2f*)(bp + s * 128);
      v2f b1n = *(const v2f*)(bp + s * 128 + 64);
      c0 = __builtin_amdgcn_wmma_f32_16x16x4_f32(
          false, a, false, b0, (short)0, c0, false, false);
      c1 = __builtin_amdgcn_wmma_f32_16x16x4_f32(
          false, a, false, b1, (short)0, c1, false, false);
      a = an; b0 = b0n; b1 = b1n;
    }
    c0 = __builtin_amdgcn_wmma_f32_16x16x4_f32(
        false, a, false, b0, (short)0, c0, false, false);
    c1 = __builtin_amdgcn_wmma_f32_16x16x4_f32(
        false, a, false, b1, (short)0, c1, false, false);

    // score = ||e||^2 - 2*dot ; running argmin (strict < keeps first index)
    const int n0 = kb * 32 + (lane & 15);
    const int n1 = n0 + 16;
    const float e0 = enorm[n0];
    const float e1 = enorm[n1];
#pragma unroll
    for (int j = 0; j < 8; ++j) {
      float s0 = fmaf(-2.f, c0[j], e0);
      if (s0 < minv[j]) { minv[j] = s0; mini[j] = n0; }
      float s1 = fmaf(-2.f, c1[j], e1);
      if (s1 < minv[j]) { minv[j] = s1; mini[j] = n1; }
    }

    wait_async0();       // next B tile fully landed in LDS
    __syncthreads();     // ... and visible to all waves; cur buffer reusable
  }

  // cross-lane argmin within each 16-lane half (lanes 0-15: rows j,
  // lanes 16-31: rows j+8 of this wave's tile), first-index tie-break
#pragma unroll
  for (int m = 8; m >= 1; m >>= 1) {
#pragma unroll
    for (int j = 0; j < 8; ++j) {
      float ov = __shfl_xor(minv[j], m, 16);
      int   oi = __shfl_xor(mini[j], m, 16);
      if (ov < minv[j] || (ov == minv[j] && oi < mini[j])) {
        minv[j] = ov;
        mini[j] = oi;
      }
    }
  }

  if ((lane & 15) == 0) {
    const int rbase = row0 + wave * 16 + ((lane >> 4) << 3);
#pragma unroll
    for (int j = 0; j < 8; ++j) out[rbase + j] = (float)mini[j];
  }
}

// ---------------------------------------------------------------------------
extern "C" void kernel_launch(void* const* d_in, const int* in_sizes, int n_in,
                              void* d_out, int out_size, void* d_ws, size_t ws_size,
                              hipStream_t stream) {
  const float* z   = (const float*)d_in[0];   // [N, 256]
  const float* emb = (const float*)d_in[1];   // [1024, 256]
  float* out   = (float*)d_out;               // [N] indices (as float)
  float* enorm = (float*)d_ws;                // [1024] scratch

  const int N = in_sizes[0] / D_DIM;          // 32768

  vq_enorm_kernel<<<K_DIM / WAVES, THREADS, 0, stream>>>(emb, enorm);

  const size_t lds_bytes = LDS_FLOATS * sizeof(float);  // 128KB
  vq_argmin_kernel<<<N / 64, THREADS, lds_bytes, stream>>>(z, emb, enorm, out);
}